// PatchNetVLAD_88046829568484
// MI455X (gfx1250) — compile-verified
//
#include <hip/hip_runtime.h>
#include <cstddef>
#include <cstdint>

// Problem constants (PatchNetVLAD reference)
#define B_   2
#define C_   512
#define H_   30
#define W_   40
#define HW_  1200
#define K_   64
#define HP_  27
#define WP_  37
#define NP_  999          // HP_*WP_
#define EPS_ 1e-12f
#define VLOCAL_ ((size_t)B_ * K_ * C_ * NP_)   // 65,470,464

typedef __attribute__((ext_vector_type(2))) float v2f;
typedef __attribute__((ext_vector_type(8))) float v8f;

static __device__ __forceinline__ v8f zero8() {
  v8f z;
#pragma unroll
  for (int i = 0; i < 8; ++i) z[i] = 0.f;
  return z;
}

// ---- CDNA5 async / TDM helpers (guarded so compile never breaks) ----------
#if __has_builtin(__builtin_amdgcn_global_load_async_to_lds_b128)
#define HAVE_ASYNC_B128 1
#endif
#if __has_builtin(__builtin_amdgcn_tensor_load_to_lds)
#define HAVE_TDM 1
#endif

// The async builtins take pointers to GCC-vector int4 in AS1 (global) and
// AS3 (LDS): parameter prints as 'int __vector(4) __device__ *'.
typedef int v4i_gcc __attribute__((__vector_size__(16)));
typedef __attribute__((address_space(1))) v4i_gcc gl_v4i;
typedef __attribute__((address_space(3))) v4i_gcc lds_v4i;

static __device__ __forceinline__ gl_v4i* as_global_v4(const void* p) {
  return (gl_v4i*)(uintptr_t)p;
}
static __device__ __forceinline__ lds_v4i* as_lds_v4(const void* p) {
  return (lds_v4i*)(uint32_t)(uintptr_t)p;   // low 32 bits = LDS offset
}

static __device__ __forceinline__ void wait_async0() {
#if __has_builtin(__builtin_amdgcn_s_wait_asynccnt)
  __builtin_amdgcn_s_wait_asynccnt(0);
#else
  asm volatile("s_wait_asynccnt 0x0" ::: "memory");
#endif
}
static __device__ __forceinline__ void wait_tensor0() {
#if __has_builtin(__builtin_amdgcn_s_wait_tensorcnt)
  __builtin_amdgcn_s_wait_tensorcnt(0);
#else
  asm volatile("s_wait_tensorcnt 0x0" ::: "memory");
#endif
}

#if defined(HAVE_TDM)
// 1-D contiguous row copy global->LDS through the Tensor Data Mover.
// D# built per CDNA5 ISA 8.3/8.4: group0 = {count=1 | lds_addr | global_addr
// | type=2}, group1 = {data_size=4B, tensor_dim0=tile_dim0=nelem, dims1=1}.
static __device__ __forceinline__ void tdm_load_row_to_lds(const float* gsrc,
                                                           void* lds_dst,
                                                           int nelem) {
  typedef unsigned int u32x4 __attribute__((ext_vector_type(4)));
  typedef int i32x4 __attribute__((ext_vector_type(4)));
  typedef int i32x8 __attribute__((ext_vector_type(8)));
  const unsigned long long ga = (unsigned long long)(uintptr_t)gsrc;
  u32x4 g0;
  g0[0] = 1u;                                          // count=1 (valid), user
  g0[1] = (unsigned int)(uintptr_t)lds_dst;            // lds_addr (bytes)
  g0[2] = (unsigned int)ga;                            // global_addr[31:0]
  g0[3] = (unsigned int)((ga >> 32) & 0x01FFFFFFu) | 0x80000000u; // [56:32]|type=2
  i32x8 g1;
  g1[0] = 0x00020000;                    // workgroup_mask=0, data_size=4B
  g1[1] = (int)((unsigned)nelem << 16);  // barrier_addr=0 | tensor_dim0 lo16
  g1[2] = (int)(1u << 16);               // tensor_dim0 hi=0 | tensor_dim1=1 lo16
  g1[3] = (int)((unsigned)nelem << 16);  // tensor_dim1 hi=0 | tile_dim0
  g1[4] = 1;                             // tile_dim1=1, tile_dim2=0
  g1[5] = nelem;                         // tensor_dim0_stride lo32
  g1[6] = (int)((unsigned)nelem << 16);  // stride0 hi=0 | tensor_dim1_stride lo16
  g1[7] = 0;
  i32x4 z4 = {0, 0, 0, 0};
#if __clang_major__ >= 23
  i32x8 z8 = {0, 0, 0, 0, 0, 0, 0, 0};
  __builtin_amdgcn_tensor_load_to_lds(g0, g1, z4, z4, z8, 0);
#else
  __builtin_amdgcn_tensor_load_to_lds(g0, g1, z4, z4, 0);
#endif
}
#endif

// ---------------------------------------------------------------------------
// K1: per-pixel L2 normalization over C. 75 blocks x 256 thr; lanes->pixels
// (coalesced per-c row), 8-way c-split reduced through LDS.
// ---------------------------------------------------------------------------
__global__ void __launch_bounds__(256) k_xnorm(const float* __restrict__ x,
                                               float* __restrict__ xn) {
  __shared__ float red[8][32];
  const int pj = threadIdx.x & 31;
  const int cs = threadIdx.x >> 5;               // 0..7
  const int t  = blockIdx.x * 32 + pj;           // pixel id in [0, B*HW)
  const int b  = t / HW_;
  const int hw = t % HW_;
  const float* xp = x  + (size_t)b * C_ * HW_ + hw;
  float ss = 0.f;
  for (int c = cs * 64; c < cs * 64 + 64; ++c) {
    float v = xp[(size_t)c * HW_];
    ss += v * v;
  }
  red[cs][pj] = ss;
  __syncthreads();
  if (cs == 0) {
    float s = 0.f;
#pragma unroll
    for (int i = 0; i < 8; ++i) s += red[i][pj];
    red[0][pj] = 1.f / fmaxf(sqrtf(s), EPS_);
  }
  __syncthreads();
  const float inv = red[0][pj];
  float* xo = xn + (size_t)b * C_ * HW_ + hw;
  for (int c = cs * 64; c < cs * 64 + 64; ++c)
    xo[(size_t)c * HW_] = xp[(size_t)c * HW_] * inv;
}

// ---------------------------------------------------------------------------
// K2: logits = conv_w (64x512) @ xn (512 x HW), tiled 16x16, inner C via
// v_wmma_f32_16x16x4_f32 (128 chained WMMAs per wave), then softmax over K.
// Block = (b, hw-tile of 16), 4 waves = 4 k-tiles.
// ---------------------------------------------------------------------------
__global__ void __launch_bounds__(128) k_logits_softmax(
    const float* __restrict__ conv_w, const float* __restrict__ xn,
    float* __restrict__ soft) {
  __shared__ float logits[K_][16];
  const int b    = blockIdx.x / (HW_ / 16);
  const int hw0  = (blockIdx.x % (HW_ / 16)) * 16;
  const int wave = threadIdx.x >> 5;             // = k-tile
  const int lane = threadIdx.x & 31;
  const int m    = lane & 15;
  const int half = lane >> 4;
  const int krow = wave * 16 + m;                // A: M = k
  const float* xb = xn + (size_t)b * C_ * HW_;

  v8f acc = zero8();
  for (int cs = 0; cs < C_; cs += 4) {
    const int cc = cs + 2 * half;                // A/B inner-K lane split
    v2f a, bf;
    a.x  = conv_w[(size_t)krow * C_ + cc];
    a.y  = conv_w[(size_t)krow * C_ + cc + 1];
    bf.x = xb[(size_t)cc * HW_ + hw0 + m];       // B: K = c, N = hw
    bf.y = xb[(size_t)(cc + 1) * HW_ + hw0 + m];
    acc = __builtin_amdgcn_wmma_f32_16x16x4_f32(false, a, false, bf,
                                                (short)0, acc, false, false);
  }
#pragma unroll
  for (int j = 0; j < 8; ++j)                    // C/D layout: M=j+8*half, N=m
    logits[wave * 16 + j + 8 * half][m] = acc[j];
  __syncthreads();

  if (threadIdx.x < 16) {
    const int col = threadIdx.x;
    float mx = -1e30f;
    for (int k = 0; k < K_; ++k) mx = fmaxf(mx, logits[k][col]);
    float s = 0.f;
    for (int k = 0; k < K_; ++k) s += expf(logits[k][col] - mx);
    const float invs = 1.f / s;
    for (int k = 0; k < K_; ++k)
      soft[((size_t)b * K_ + k) * HW_ + hw0 + col] =
          expf(logits[k][col] - mx) * invs;
  }
}

// ---------------------------------------------------------------------------
// K3: vlad_global. G[k,c] = sum_hw soft*xn - centroid*S  (GEMM, inner HW=1200
// via WMMA f32 16x16x4), then both L2 norms fused in-block (1 block per b,
// 16 waves = 4 k-tiles x 4 c-groups of 8 c-tiles).
// ---------------------------------------------------------------------------
__global__ void __launch_bounds__(512) k_vlad_global(
    const float* __restrict__ soft, const float* __restrict__ xn,
    const float* __restrict__ centroids, float* __restrict__ out) {
  __shared__ float Ssh[K_];
  __shared__ float n1s[K_];
  __shared__ float inv2_sh;
  const int b   = blockIdx.x;
  const int tid = threadIdx.x;

  if (tid < K_) { Ssh[tid] = 0.f; n1s[tid] = 0.f; }
  __syncthreads();
  {                                              // S[b,k] = sum_hw soft, 8-way
    const int kk = tid >> 3, seg = tid & 7;
    const float* sp2 = soft + ((size_t)b * K_ + kk) * HW_ + seg * (HW_ / 8);
    float s = 0.f;
    for (int i = 0; i < HW_ / 8; ++i) s += sp2[i];
    atomicAdd(&Ssh[kk], s);
  }
  __syncthreads();

  const int wave = tid >> 5, lane = tid & 31;
  const int kt = wave & 3, cg = wave >> 2;
  const int m = lane & 15, half = lane >> 4;
  const int krow = kt * 16 + m;
  const float* sb = soft + ((size_t)b * K_ + krow) * HW_;
  const float* xb = xn + (size_t)b * C_ * HW_;

  v8f acc[8];
#pragma unroll
  for (int i = 0; i < 8; ++i) acc[i] = zero8();

  for (int hw = 0; hw < HW_; hw += 4) {
    const int hh = hw + 2 * half;
    v2f a;
    a.x = sb[hh];
    a.y = sb[hh + 1];
#pragma unroll
    for (int i = 0; i < 8; ++i) {
      const int c = (cg * 8 + i) * 16 + m;
      v2f bf;
      bf.x = xb[(size_t)c * HW_ + hh];
      bf.y = xb[(size_t)c * HW_ + hh + 1];
      acc[i] = __builtin_amdgcn_wmma_f32_16x16x4_f32(false, a, false, bf,
                                                     (short)0, acc[i],
                                                     false, false);
    }
  }

  // per-k sum of squares of G (after centroid subtraction)
#pragma unroll
  for (int j = 0; j < 8; ++j) {
    const int kj = kt * 16 + j + 8 * half;
    const float sj = Ssh[kj];
    float loc = 0.f;
#pragma unroll
    for (int i = 0; i < 8; ++i) {
      const int c = (cg * 8 + i) * 16 + m;
      const float g = acc[i][j] - centroids[(size_t)kj * C_ + c] * sj;
      loc += g * g;
    }
    loc += __shfl_xor(loc, 8, 32);               // reduce over 16-lane half
    loc += __shfl_xor(loc, 4, 32);
    loc += __shfl_xor(loc, 2, 32);
    loc += __shfl_xor(loc, 1, 32);
    if (m == 0) atomicAdd(&n1s[kj], loc);
  }
  __syncthreads();

  if (tid == 0) {
    float s2 = 0.f;
    for (int k = 0; k < K_; ++k) {
      const float nk = sqrtf(n1s[k]);
      const float r = nk / fmaxf(nk, EPS_);
      s2 += r * r;
      n1s[k] = 1.f / fmaxf(nk, EPS_);
    }
    inv2_sh = 1.f / fmaxf(sqrtf(s2), EPS_);
  }
  __syncthreads();
  const float inv2 = inv2_sh;

#pragma unroll
  for (int j = 0; j < 8; ++j) {
    const int kj = kt * 16 + j + 8 * half;
    const float sj = Ssh[kj];
    const float f = n1s[kj] * inv2;
#pragma unroll
    for (int i = 0; i < 8; ++i) {
      const int c = (cg * 8 + i) * 16 + m;
      const float g = acc[i][j] - centroids[(size_t)kj * C_ + c] * sj;
      out[VLOCAL_ + ((size_t)b * K_ + kj) * C_ + c] = g * f;
    }
  }
}

// ---------------------------------------------------------------------------
// K4a: zero the global n1^2 accumulator.
// ---------------------------------------------------------------------------
__global__ void k_zero(float* __restrict__ p, int n) {
  const int i = blockIdx.x * 256 + threadIdx.x;
  if (i < n) p[i] = 0.f;
}

// ---------------------------------------------------------------------------
// K4: main 262MB-producing pass. Block = (b, k, c-chunk of 128) -> 512 blocks.
// box[b,k,c,p] = (sum_{4x4} soft*xn - centroid[k,c]*Sbox[b,k,p]) / 16
// soft row staged to LDS via TDM (tensor_load_to_lds), centroids slice via
// global_load_async_to_lds_b128; lanes map to consecutive p -> coalesced
// stores/loads; C-norm accumulated via ds_add_f32 then one global atomic per p.
// ---------------------------------------------------------------------------
__global__ void __launch_bounds__(256) k_local_box(
    const float* __restrict__ soft, const float* __restrict__ xn,
    const float* __restrict__ centroids, float* __restrict__ out,
    float* __restrict__ n1sq) {
  __shared__ __align__(16) float sp[HW_];
  __shared__ __align__(16) float cwl[128];
  __shared__ float sbx[NP_];
  __shared__ float n1l[NP_];
  const int cc = blockIdx.x & 3;                 // c-chunk (128 c's)
  const int k  = (blockIdx.x >> 2) & (K_ - 1);
  const int b  = blockIdx.x >> 8;
  const int tid = threadIdx.x;
  const int c0 = cc * 128;

  const float* srow = soft + ((size_t)b * K_ + k) * HW_;
  const float* crow = centroids + (size_t)k * C_ + c0;

#if defined(HAVE_TDM)
  if (tid == 0) tdm_load_row_to_lds(srow, sp, HW_);   // TDM DMA, TENSORcnt
#else
  for (int i = tid; i < HW_; i += 256) sp[i] = srow[i];
#endif
#if defined(HAVE_ASYNC_B128)
  if (tid < 32)                                       // 32 x b128 = 512B
    __builtin_amdgcn_global_load_async_to_lds_b128(
        as_global_v4(crow + tid * 4), as_lds_v4(&cwl[tid * 4]), 0, 0);
#else
  if (tid < 128) cwl[tid] = crow[tid];
#endif
  for (int i = tid; i < NP_; i += 256) n1l[i] = 0.f;
  if (tid < 32) {
#if defined(HAVE_TDM)
    wait_tensor0();
#endif
#if defined(HAVE_ASYNC_B128)
    wait_async0();
#endif
  }
  __syncthreads();

  for (int p = tid; p < NP_; p += 256) {         // Sbox for this (b,k)
    const int py = p / WP_, px = p % WP_;
    float s = 0.f;
#pragma unroll
    for (int dy = 0; dy < 4; ++dy) {
      const float* r = sp + (py + dy) * W_ + px;
      s += r[0] + r[1] + r[2] + r[3];
    }
    sbx[p] = s;
  }
  __syncthreads();

  const float* xb = xn + (size_t)b * C_ * HW_;
  float* ob = out + ((size_t)b * K_ + k) * (size_t)C_ * NP_;

  for (int idx = tid; idx < 128 * NP_; idx += 256) {
    const int ci = idx / NP_;
    const int c  = c0 + ci;
    const int p  = idx % NP_;                    // lanes -> consecutive p
    const int py = p / WP_, px = p % WP_;
    const float* xr = xb + (size_t)c * HW_ + py * W_ + px;
    __builtin_prefetch(xr + 2 * W_ + 256, 0, 3); // global_prefetch_b8
    float g = 0.f;
#pragma unroll
    for (int dy = 0; dy < 4; ++dy) {
      const float* r  = sp + (py + dy) * W_ + px;
      const float* xq = xr + dy * W_;
      g += r[0] * xq[0] + r[1] * xq[1] + r[2] * xq[2] + r[3] * xq[3];
    }
    g = (g - cwl[ci] * sbx[p]) * 0.0625f;
    ob[(size_t)c * NP_ + p] = g;                 // coalesced 262MB write
    atomicAdd(&n1l[p], g * g);                   // ds_add_f32, conflict-free
  }
  __syncthreads();

  float* no = n1sq + ((size_t)b * K_ + k) * NP_;
  for (int p = tid; p < NP_; p += 256) atomicAdd(&no[p], n1l[p]);
}

// ---------------------------------------------------------------------------
// K5: combine norms. n2 over k*c collapses to sqrt(sum_k (n1/max(n1,eps))^2)
// because the C-normalized rows are unit vectors. invn = 1/(max(n1)*max(n2)).
// ---------------------------------------------------------------------------
__global__ void k_invn(const float* __restrict__ n1sq,
                       float* __restrict__ invn) {
  const int t = blockIdx.x * 256 + threadIdx.x;
  if (t >= B_ * NP_) return;
  const int b = t / NP_, p = t % NP_;
  float s2 = 0.f;
  for (int k = 0; k < K_; ++k) {
    const float n = sqrtf(n1sq[((size_t)b * K_ + k) * NP_ + p]);
    const float r = n / fmaxf(n, EPS_);
    s2 += r * r;
  }
  const float inv2 = 1.f / fmaxf(sqrtf(s2), EPS_);
  for (int k = 0; k < K_; ++k) {
    const float n = sqrtf(n1sq[((size_t)b * K_ + k) * NP_ + p]);
    invn[((size_t)b * K_ + k) * NP_ + p] = inv2 / fmaxf(n, EPS_);
  }
}

// ---------------------------------------------------------------------------
// K6: final rescale of vlad_local, float4 (b128) grid-stride, coalesced R+W.
// VLOCAL_ is divisible by 4 and the region starts at d_out -> 16B aligned.
// ---------------------------------------------------------------------------
__global__ void __launch_bounds__(256) k_rescale(float* __restrict__ out,
                                                 const float* __restrict__ invn) {
  const size_t n4 = VLOCAL_ / 4;
  const size_t stride = (size_t)gridDim.x * 256;
  float4* o4 = (float4*)out;
  for (size_t q = (size_t)blockIdx.x * 256 + threadIdx.x; q < n4; q += stride) {
    float4 v = o4[q];
    const unsigned int i0 = (unsigned int)(q * 4);
    float f[4];
#pragma unroll
    for (int j = 0; j < 4; ++j) {
      const unsigned int i = i0 + j;
      const unsigned int kcp = i / NP_;          // (b*K+k)*C + c
      const unsigned int p = i - kcp * NP_;
      f[j] = invn[(size_t)(kcp >> 9) * NP_ + p];
    }
    v.x *= f[0]; v.y *= f[1]; v.z *= f[2]; v.w *= f[3];
    o4[q] = v;
  }
}

// ---------------------------------------------------------------------------
extern "C" void kernel_launch(void* const* d_in, const int* in_sizes, int n_in,
                              void* d_out, int out_size, void* d_ws,
                              size_t ws_size, hipStream_t stream) {
  (void)in_sizes; (void)n_in; (void)out_size; (void)ws_size;
  const float* x         = (const float*)d_in[0];
  const float* conv_w    = (const float*)d_in[1];
  const float* centroids = (const float*)d_in[2];
  float* out = (float*)d_out;

  // workspace layout (floats): xn | soft | n1sq | invn  (~6.6 MB total)
  float* xn   = (float*)d_ws;
  float* soft = xn + (size_t)B_ * C_ * HW_;      // 1,228,800
  float* n1sq = soft + (size_t)B_ * K_ * HW_;    //   153,600
  float* invn = n1sq + (size_t)B_ * K_ * NP_;    //   127,872 (+127,872)

  k_xnorm<<<(B_ * HW_) / 32, 256, 0, stream>>>(x, xn);
  k_logits_softmax<<<B_ * (HW_ / 16), 128, 0, stream>>>(conv_w, xn, soft);
  k_vlad_global<<<B_, 512, 0, stream>>>(soft, xn, centroids, out);
  k_zero<<<(B_ * K_ * NP_ + 255) / 256, 256, 0, stream>>>(n1sq, B_ * K_ * NP_);
  k_local_box<<<B_ * K_ * 4, 256, 0, stream>>>(soft, xn, centroids, out, n1sq);
  k_invn<<<(B_ * NP_ + 255) / 256, 256, 0, stream>>>(n1sq, invn);
  k_rescale<<<4096, 256, 0, stream>>>(out, invn);
}